// AdaptiveAttention_80642305950049
// MI455X (gfx1250) — compile-verified
//
#include <hip/hip_runtime.h>
#include <math.h>

// ---------------------------------------------------------------------------
// AdaptiveAttention for MI455X (gfx1250, wave32).
// Memory-bound: ~768MB HBM traffic total -> ~33us at 23.3 TB/s.
// WMMA (V_WMMA_F32_16X16X4_F32, exact fp32) used for the shared-FC GEMMs.
// ---------------------------------------------------------------------------

typedef __attribute__((ext_vector_type(2))) float v2f;
typedef __attribute__((ext_vector_type(8))) float v8f;

#define B_   16
#define C_   256
#define HW_  16384
#define H_   128
#define W_   128
#define HID_ 16
#define SBLK 16                  // spatial blocks per batch image
#define PIXB (HW_ / SBLK)        // 1024 pixels per block
#define NPART 128                // SBLK * 8 waves per block
#define NEG_INF (-3.402823466e38f)

__device__ __forceinline__ float sigmoidf_(float v) {
  return 1.0f / (1.0f + __expf(-v));
}

// ---------------------------------------------------------------------------
// Pass 1: single read of x producing ALL reductions.
//   per (b,c): spatial sum / max   -> 128 deterministic wave partials
//   per (b,hw): channel sum / max  -> csum / cmax    (spatial-attn input)
//   per (b,hw): sum_c gate_w[c]*x  -> gsum           (gate pre-activation)
// grid = B*SBLK blocks x 256 threads; thread owns 4 contiguous pixels (float4)
// ---------------------------------------------------------------------------
__global__ void reduce_pass_kernel(const float* __restrict__ x,
                                   const float* __restrict__ gate_w,
                                   float* __restrict__ part_sum,
                                   float* __restrict__ part_max,
                                   float* __restrict__ csum,
                                   float* __restrict__ cmax,
                                   float* __restrict__ gsum) {
  const int blk  = blockIdx.x & (SBLK - 1);
  const int b    = blockIdx.x / SBLK;
  const int tid  = threadIdx.x;            // 0..255
  const int wave = tid >> 5;               // 0..7 (wave32)
  const int lane = tid & 31;
  const int pix  = blk * PIXB + tid * 4;   // 4 contiguous pixels per thread
  const float* xb = x + (size_t)b * C_ * HW_ + pix;

  float4 cs = make_float4(0.f, 0.f, 0.f, 0.f);
  float4 cm = make_float4(NEG_INF, NEG_INF, NEG_INF, NEG_INF);
  float4 gs = make_float4(0.f, 0.f, 0.f, 0.f);

  for (int c = 0; c < C_; ++c) {
    const float4 v = *(const float4*)(xb + (size_t)c * HW_);
    const float gw = gate_w[c];            // uniform scalar load (L2/K$)

    cs.x += v.x; cs.y += v.y; cs.z += v.z; cs.w += v.w;
    cm.x = fmaxf(cm.x, v.x); cm.y = fmaxf(cm.y, v.y);
    cm.z = fmaxf(cm.z, v.z); cm.w = fmaxf(cm.w, v.w);
    gs.x += gw * v.x; gs.y += gw * v.y; gs.z += gw * v.z; gs.w += gw * v.w;

    // wave32 spatial reduction for this channel
    float s = v.x + v.y + v.z + v.w;
    float m = fmaxf(fmaxf(v.x, v.y), fmaxf(v.z, v.w));
    #pragma unroll
    for (int off = 16; off > 0; off >>= 1) {
      s += __shfl_xor(s, off);
      m = fmaxf(m, __shfl_xor(m, off));
    }
    if (lane == 0) {
      const int idx = (b * C_ + c) * NPART + blk * 8 + wave;
      part_sum[idx] = s;
      part_max[idx] = m;
    }
  }

  const int o = b * HW_ + pix;
  *(float4*)(csum + o) = cs;
  *(float4*)(cmax + o) = cm;
  *(float4*)(gsum + o) = gs;
}

// ---------------------------------------------------------------------------
// Combine the 128 wave partials per (b,c) -> sum_bc / max_bc. Deterministic.
// ---------------------------------------------------------------------------
__global__ void combine_partials_kernel(const float* __restrict__ part_sum,
                                        const float* __restrict__ part_max,
                                        float* __restrict__ sum_bc,
                                        float* __restrict__ max_bc) {
  const int gid = blockIdx.x * blockDim.x + threadIdx.x;  // 0..B_*C_-1
  if (gid >= B_ * C_) return;
  float s = 0.f, m = NEG_INF;
  const float* ps = part_sum + (size_t)gid * NPART;
  const float* pm = part_max + (size_t)gid * NPART;
  for (int i = 0; i < NPART; ++i) {
    s += ps[i];
    m = fmaxf(m, pm[i]);
  }
  sum_bc[gid] = s;
  max_bc[gid] = m;
}

// ---------------------------------------------------------------------------
// Shared FC stack on the WMMA unit (one wave does everything).
//   FC1: [16,256]x[256,16] via 64x V_WMMA_F32_16X16X4_F32 (exact fp32)
//   ACON-C activation, relayout through LDS (D-layout -> A-layout)
//   FC2: 16 tiles of [16,16] K=16, accumulate avg-path + max-path
//   fused BatchNorm (affine) + sigmoid -> ca[16][256]
// A 16x4 f32 operand layout (ISA 7.12.2): lane<16: M=lane, VGPR0/1 = K0/K1;
// lane>=16: K2/K3.  B 4x16 mirrors with N=lane&15.  C/D: VGPR r holds
// (M = r + 8*(lane/16), N = lane&15).
// ---------------------------------------------------------------------------
__global__ void fc_wmma_kernel(const float* __restrict__ sum_bc,
                               const float* __restrict__ max_bc,
                               const float* __restrict__ w1,
                               const float* __restrict__ p1,
                               const float* __restrict__ p2,
                               const float* __restrict__ aconb,
                               const float* __restrict__ w2,
                               const float* __restrict__ bn_gamma,
                               const float* __restrict__ bn_beta,
                               const float* __restrict__ bn_mean,
                               const float* __restrict__ bn_var,
                               float* __restrict__ ca) {
  __shared__ float hA[HID_ * HID_];   // ACON(avg path) [16][16]
  __shared__ float hM[HID_ * HID_];   // ACON(max path) [16][16]

  const int lane  = threadIdx.x;      // 0..31, full wave, EXEC all ones
  const int half  = lane >> 4;        // 0/1
  const int idx16 = lane & 15;        // M for A-operand, N for B-operand
  const float inv_hw = 1.0f / (float)HW_;

  // ---- FC1 + ACON for both pooled inputs ----
  for (int pass = 0; pass < 2; ++pass) {
    const float* src = pass ? max_bc : sum_bc;
    const float scale = pass ? 1.0f : inv_hw;   // avg pool = sum/HW
    v8f h = {};
    for (int k0 = 0; k0 < C_; k0 += 4) {
      const int k = k0 + 2 * half;
      v2f av, bv;
      av.x = src[idx16 * C_ + k]     * scale;   // A[m][k],   m = idx16
      av.y = src[idx16 * C_ + k + 1] * scale;
      bv.x = w1[idx16 * C_ + k];                // B[k][n] = w1[n][k]
      bv.y = w1[idx16 * C_ + k + 1];
      h = __builtin_amdgcn_wmma_f32_16x16x4_f32(
              false, av, false, bv, (short)0, h, false, false);
    }
    // ACON-C: d = (p1-p2)*h;  act = d*sigmoid(beta*d) + p2*h   (j = idx16)
    const float P1 = p1[idx16], P2 = p2[idx16], AB = aconb[idx16];
    float* dst = pass ? hM : hA;
    #pragma unroll
    for (int r = 0; r < 8; ++r) {
      const float hv = h[r];
      const float d  = (P1 - P2) * hv;
      const float act = d * sigmoidf_(AB * d) + P2 * hv;
      const int m = r + 8 * half;                // D-layout row
      dst[m * HID_ + idx16] = act;
    }
  }
  __syncthreads();

  // ---- FC2 (both paths summed) + fused BN + sigmoid ----
  for (int t = 0; t < 16; ++t) {
    const int cidx = t * 16 + idx16;             // global channel (N)
    v8f o = {};
    for (int srci = 0; srci < 2; ++srci) {
      const float* hh = srci ? hM : hA;
      #pragma unroll
      for (int k0 = 0; k0 < HID_; k0 += 4) {
        const int k = k0 + 2 * half;
        v2f av, bv;
        av.x = hh[idx16 * HID_ + k];             // A[m][k], m = idx16
        av.y = hh[idx16 * HID_ + k + 1];
        bv.x = w2[cidx * HID_ + k];              // B[k][n] = w2[c][k]
        bv.y = w2[cidx * HID_ + k + 1];
        o = __builtin_amdgcn_wmma_f32_16x16x4_f32(
                false, av, false, bv, (short)0, o, false, false);
      }
    }
    // BN is affine: BN(o_avg)+BN(o_mx) = (o_avg+o_mx)*s + 2*(beta - mean*s)
    const float s  = bn_gamma[cidx] * rsqrtf(bn_var[cidx] + 1e-5f);
    const float t2 = 2.0f * (bn_beta[cidx] - bn_mean[cidx] * s);
    #pragma unroll
    for (int r = 0; r < 8; ++r) {
      const int m = r + 8 * half;
      ca[m * C_ + cidx] = sigmoidf_(o[r] * s + t2);
    }
  }
}

// ---------------------------------------------------------------------------
// Spatial attention: 7x7 conv (pad 3) over [mean_c, max_c], then sigmoid.
// Inputs are 1MB arrays -> L2-resident, compute is trivial.
// ---------------------------------------------------------------------------
__global__ void spatial_conv_kernel(const float* __restrict__ csum,
                                    const float* __restrict__ cmax,
                                    const float* __restrict__ sa_w,
                                    const float* __restrict__ sa_b,
                                    float* __restrict__ sig_sc) {
  const int gid = blockIdx.x * blockDim.x + threadIdx.x;  // 0..B*HW-1
  if (gid >= B_ * HW_) return;
  const int b  = gid >> 14;
  const int hw = gid & (HW_ - 1);
  const int h  = hw >> 7;
  const int w  = hw & (W_ - 1);
  const float inv_c = 1.0f / (float)C_;

  float acc = sa_b[0];
  #pragma unroll
  for (int kh = 0; kh < 7; ++kh) {
    const int ph = h + kh - 3;
    if (ph < 0 || ph >= H_) continue;
    #pragma unroll
    for (int kw = 0; kw < 7; ++kw) {
      const int pw = w + kw - 3;
      if (pw < 0 || pw >= W_) continue;
      const int pidx = b * HW_ + ph * W_ + pw;
      acc += sa_w[kh * 7 + kw]      * (csum[pidx] * inv_c)  // in-ch 0: mean
           + sa_w[49 + kh * 7 + kw] * cmax[pidx];            // in-ch 1: max
    }
  }
  sig_sc[gid] = sigmoidf_(acc);
}

// ---------------------------------------------------------------------------
// Final fused elementwise pass: read x once, write out once (float4).
//   ch  = x*ca + 0.1*(x - avg)
//   sp  = x*sigmoid(sc)
//   g   = sigmoid(gsum + gate_b)
//   out = x + g*(alpha*ch + beta*sp)
// ---------------------------------------------------------------------------
__global__ void final_kernel(const float* __restrict__ x,
                             const float* __restrict__ ca,
                             const float* __restrict__ sum_bc,
                             const float* __restrict__ sig_sc,
                             const float* __restrict__ gsum,
                             const float* __restrict__ gate_b,
                             const float* __restrict__ alpha_p,
                             const float* __restrict__ beta_p,
                             float* __restrict__ out) {
  const long i4 = (long)blockIdx.x * blockDim.x + threadIdx.x;
  const long total4 = (long)B_ * C_ * HW_ / 4;
  if (i4 >= total4) return;

  const long e  = i4 * 4;
  const int hw  = (int)(e & (HW_ - 1));   // HW_ divisible by 4 -> shared (b,c)
  const int bc  = (int)(e >> 14);         // b*C_ + c
  const int b   = bc >> 8;

  const float cav  = ca[bc];
  const float avgv = sum_bc[bc] * (1.0f / (float)HW_);
  const float gb = gate_b[0];
  const float al = alpha_p[0];
  const float be = beta_p[0];

  const float4 xv  = ((const float4*)x)[i4];
  const float4 sav = *(const float4*)(sig_sc + (size_t)b * HW_ + hw);
  const float4 gsv = *(const float4*)(gsum   + (size_t)b * HW_ + hw);

  float4 ov;
  {
    const float xx = xv.x, g = sigmoidf_(gsv.x + gb);
    ov.x = xx + g * (al * (xx * cav + 0.1f * (xx - avgv)) + be * (xx * sav.x));
  }
  {
    const float xx = xv.y, g = sigmoidf_(gsv.y + gb);
    ov.y = xx + g * (al * (xx * cav + 0.1f * (xx - avgv)) + be * (xx * sav.y));
  }
  {
    const float xx = xv.z, g = sigmoidf_(gsv.z + gb);
    ov.z = xx + g * (al * (xx * cav + 0.1f * (xx - avgv)) + be * (xx * sav.z));
  }
  {
    const float xx = xv.w, g = sigmoidf_(gsv.w + gb);
    ov.w = xx + g * (al * (xx * cav + 0.1f * (xx - avgv)) + be * (xx * sav.w));
  }
  ((float4*)out)[i4] = ov;
}

// ---------------------------------------------------------------------------
extern "C" void kernel_launch(void* const* d_in, const int* in_sizes, int n_in,
                              void* d_out, int out_size, void* d_ws, size_t ws_size,
                              hipStream_t stream) {
  const float* x      = (const float*)d_in[0];
  const float* w1     = (const float*)d_in[1];
  const float* p1     = (const float*)d_in[2];
  const float* p2     = (const float*)d_in[3];
  const float* aconb  = (const float*)d_in[4];
  const float* w2     = (const float*)d_in[5];
  const float* bn_g   = (const float*)d_in[6];
  const float* bn_b   = (const float*)d_in[7];
  const float* bn_m   = (const float*)d_in[8];
  const float* bn_v   = (const float*)d_in[9];
  const float* sa_w   = (const float*)d_in[10];
  const float* sa_b   = (const float*)d_in[11];
  const float* gate_w = (const float*)d_in[12];
  const float* gate_b = (const float*)d_in[13];
  const float* alpha  = (const float*)d_in[14];
  const float* beta   = (const float*)d_in[15];
  float* out = (float*)d_out;

  // Workspace layout (floats). Total ~8.4 MB.
  float* ws       = (float*)d_ws;
  float* part_sum = ws;                                 // B*C*128
  float* part_max = part_sum + (size_t)B_ * C_ * NPART; // B*C*128
  float* csum     = part_max + (size_t)B_ * C_ * NPART; // B*HW
  float* cmax     = csum + (size_t)B_ * HW_;            // B*HW
  float* gsum     = cmax + (size_t)B_ * HW_;            // B*HW
  float* sum_bc   = gsum + (size_t)B_ * HW_;            // B*C
  float* max_bc   = sum_bc + B_ * C_;                   // B*C
  float* ca       = max_bc + B_ * C_;                   // B*C
  float* sig_sc   = ca + B_ * C_;                       // B*HW

  reduce_pass_kernel<<<B_ * SBLK, 256, 0, stream>>>(
      x, gate_w, part_sum, part_max, csum, cmax, gsum);

  combine_partials_kernel<<<(B_ * C_) / 256, 256, 0, stream>>>(
      part_sum, part_max, sum_bc, max_bc);

  fc_wmma_kernel<<<1, 32, 0, stream>>>(
      sum_bc, max_bc, w1, p1, p2, aconb, w2, bn_g, bn_b, bn_m, bn_v, ca);

  spatial_conv_kernel<<<(B_ * HW_) / 256, 256, 0, stream>>>(
      csum, cmax, sa_w, sa_b, sig_sc);

  final_kernel<<<((long)B_ * C_ * HW_ / 4) / 256, 256, 0, stream>>>(
      x, ca, sum_bc, sig_sc, gsum, gate_b, alpha, beta, out);
}